// ResBlock_6682969112862
// MI455X (gfx1250) — compile-verified
//
#include <hip/hip_runtime.h>
#include <hip/hip_bf16.h>

#define DD 128
#define BN_EPS 1e-5f

typedef float v2f __attribute__((ext_vector_type(2)));
typedef float v8f __attribute__((ext_vector_type(8)));

// ---------------- init: deg=1 (self loops), zero column stats ----------------
__global__ __launch_bounds__(256)
void k_init(int* __restrict__ deg, float* __restrict__ colsum,
            float* __restrict__ colsq, int N) {
    int i = blockIdx.x * 256 + threadIdx.x;
    if (i < N) deg[i] = 1;
    if (i < DD) { colsum[i] = 0.0f; colsq[i] = 0.0f; }
}

// ---------------- degree: count incoming edges ----------------
__global__ __launch_bounds__(256)
void k_degree(const int* __restrict__ dst, int* __restrict__ deg, int E) {
    int e = blockIdx.x * 256 + threadIdx.x;
    if (e < E) atomicAdd(&deg[dst[e]], 1);
}

// ---------------- dinv = rsqrt(deg) ----------------
__global__ __launch_bounds__(256)
void k_dinv(const int* __restrict__ deg, float* __restrict__ dinv, int N) {
    int i = blockIdx.x * 256 + threadIdx.x;
    if (i < N) dinv[i] = rsqrtf((float)deg[i]);
}

// ---------------- GEMM + fused pre-scale epilogue --------------------------
// h = x @ W via V_WMMA_F32_16X16X4_F32.
// Block = 256 threads = 8 waves; wave w owns columns [16w,16w+16);
// block covers 32 rows (grid = N/32 = 3125, exact, EXEC all-1s everywhere).
// Epilogue writes g = h*dinv[row] (workspace, edge messages) and
// agg = h*dinv[row]^2 (d_out, the self-loop contribution).
__global__ __launch_bounds__(256)
void k_gemm_wmma(const float* __restrict__ x, const float* __restrict__ W,
                 const float* __restrict__ dinv,
                 float* __restrict__ g, float* __restrict__ agg) {
    const int lane = threadIdx.x & 31;
    const int wave = threadIdx.x >> 5;
    const int col  = wave * 16 + (lane & 15);   // N index this lane carries
    const int koff = (lane >> 4) * 2;           // K offset from lane half
    const int r0   = blockIdx.x * 32;

    // Preload B: W[k][col] for all 32 K-blocks (k = 4*kb + koff, +1)
    v2f B[32];
#pragma unroll
    for (int kb = 0; kb < 32; ++kb) {
        int k = kb * 4 + koff;
        B[kb].x = W[k * DD + col];
        B[kb].y = W[(k + 1) * DD + col];
    }

#pragma unroll
    for (int rt = 0; rt < 2; ++rt) {
        const int rowA = r0 + rt * 16 + (lane & 15);       // A row this lane carries
        const float* xr = x + (size_t)rowA * DD;
        v8f c = {};
#pragma unroll
        for (int kb = 0; kb < 32; ++kb) {
            v2f a = *(const v2f*)(xr + kb * 4 + koff);     // 8B-aligned b64 load
            c = __builtin_amdgcn_wmma_f32_16x16x4_f32(
                    false, a, false, B[kb], (short)0, c, false, false);
        }
        // D layout: VGPR v -> M = v + 8*(lane>=16), N = lane&15
        const int mbase = r0 + rt * 16 + ((lane >> 4) << 3);
        float* gp = g   + (size_t)mbase * DD + col;
        float* ap = agg + (size_t)mbase * DD + col;
#pragma unroll
        for (int v = 0; v < 8; ++v) {
            float dv = dinv[mbase + v];        // broadcast across lanes 0-15/16-31
            float gv = c[v] * dv;
            gp[(size_t)v * DD] = gv;           // g  = h * dinv
            ap[(size_t)v * DD] = gv * dv;      // agg = h * dinv^2 (self loop)
        }
    }
}

// ---------------- edge scatter: agg[dst] += g[src] * dinv[dst] --------------
// 32 lanes per edge, 4 columns per lane (float4 read, 4 f32 atomics).
__global__ __launch_bounds__(256)
void k_edges(const int* __restrict__ src, const int* __restrict__ dst,
             const float* __restrict__ g, const float* __restrict__ dinv,
             float* __restrict__ agg, int E) {
    int gid = blockIdx.x * 256 + threadIdx.x;
    int e = gid >> 5;
    if (e >= E) return;
    int s = src[e], d = dst[e];
    float coef = dinv[d];                       // g already carries dinv[src]
    int c0 = (gid & 31) * 4;
    const float4 hv = *(const float4*)(g + (size_t)s * DD + c0);
    float* ap = agg + (size_t)d * DD + c0;
    unsafeAtomicAdd(ap + 0, hv.x * coef);
    unsafeAtomicAdd(ap + 1, hv.y * coef);
    unsafeAtomicAdd(ap + 2, hv.z * coef);
    unsafeAtomicAdd(ap + 3, hv.w * coef);
}

// ---------------- column stats: sum and sum of squares per column ------------
__global__ __launch_bounds__(128)
void k_stats(const float* __restrict__ agg, float* __restrict__ colsum,
             float* __restrict__ colsq, int N) {
    int col = threadIdx.x;            // 128 threads = 128 columns, coalesced rows
    float s = 0.0f, sq = 0.0f;
    for (int row = blockIdx.x; row < N; row += gridDim.x) {
        float v = agg[(size_t)row * DD + col];
        s += v; sq += v * v;
    }
    unsafeAtomicAdd(&colsum[col], s);
    unsafeAtomicAdd(&colsq[col], sq);
}

// ---------------- final: BN(+b cancels) + ReLU + residual, in-place on agg ---
__global__ __launch_bounds__(256)
void k_final(float* __restrict__ agg, const float* __restrict__ x,
             const float* __restrict__ colsum, const float* __restrict__ colsq,
             const float* __restrict__ gamma, const float* __restrict__ beta,
             int N) {
    int idx = blockIdx.x * 256 + threadIdx.x;
    if (idx >= N * DD) return;
    int col = idx & (DD - 1);
    float invN = 1.0f / (float)N;
    float mean = colsum[col] * invN;
    float var  = colsq[col] * invN - mean * mean;
    float rstd = rsqrtf(var + BN_EPS);
    float v = (agg[idx] - mean) * rstd * gamma[col] + beta[col];
    v = fmaxf(v, 0.0f);
    agg[idx] = v + x[idx];
}

extern "C" void kernel_launch(void* const* d_in, const int* in_sizes, int n_in,
                              void* d_out, int out_size, void* d_ws, size_t ws_size,
                              hipStream_t stream) {
    const float* x     = (const float*)d_in[0];
    const int*   ei    = (const int*)d_in[1];   // [2, E]: src row then dst row
    const float* W     = (const float*)d_in[2];
    // d_in[3] = b : cancels exactly inside batch-norm (out-mean removes it)
    const float* gamma = (const float*)d_in[4];
    const float* beta  = (const float*)d_in[5];

    const int N = in_sizes[0] / DD;
    const int E = in_sizes[1] / 2;
    const int* src = ei;
    const int* dst = ei + E;

    // workspace layout
    float* g      = (float*)d_ws;                 // N*DD floats (h * dinv)
    int*   deg    = (int*)(g + (size_t)N * DD);   // N ints
    float* dinv   = (float*)(deg + N);            // N floats
    float* colsum = dinv + N;                     // DD floats
    float* colsq  = colsum + DD;                  // DD floats

    float* agg = (float*)d_out;                   // fully overwritten by GEMM epilogue

    const int nb_n  = (N + 255) / 256;
    const int nb_nd = (N * DD + 255) / 256;

    k_init  <<<nb_n, 256, 0, stream>>>(deg, colsum, colsq, N);
    k_degree<<<(E + 255) / 256, 256, 0, stream>>>(dst, deg, E);
    k_dinv  <<<nb_n, 256, 0, stream>>>(deg, dinv, N);
    k_gemm_wmma<<<N / 32, 256, 0, stream>>>(x, W, dinv, g, agg);
    k_edges <<<(E * 32 + 255) / 256, 256, 0, stream>>>(src, dst, g, dinv, agg, E);
    k_stats <<<256, 128, 0, stream>>>(agg, colsum, colsq, N);
    k_final <<<nb_nd, 256, 0, stream>>>(agg, x, colsum, colsq, gamma, beta, N);
}